// ScaledDotProductAttention_83296595738808
// MI455X (gfx1250) — compile-verified
//
#include <hip/hip_runtime.h>

#define N_Q  8192
#define M_K  16384
#define DIM  64
#define KSEL 160           // topk(10) * HEAD_NUM(16)
#define INV_SCALE 0.125f   // 1/SCALE

typedef float v2f __attribute__((ext_vector_type(2)));
typedef float v8f __attribute__((ext_vector_type(8)));

// Monotonic float -> uint key (order-preserving) and inverse.
__device__ __forceinline__ unsigned int f2key(float f) {
    unsigned int b = __float_as_uint(f);
    return (b & 0x80000000u) ? ~b : (b | 0x80000000u);
}
__device__ __forceinline__ float key2f(unsigned int k) {
    unsigned int b = (k & 0x80000000u) ? (k & 0x7FFFFFFFu) : ~k;
    return __uint_as_float(b);
}

// ---------------------------------------------------------------------------
// Kernel 0: init workspace + output accumulation region
// ---------------------------------------------------------------------------
__global__ __launch_bounds__(256) void init_kernel(unsigned int* __restrict__ rowmax_key,
                                                   float* __restrict__ rowsum,
                                                   float* __restrict__ out) {
    int i = blockIdx.x * 256 + threadIdx.x;
    if (i < N_Q) { rowmax_key[i] = 0u; rowsum[i] = 0.0f; }
    if (i < N_Q * DIM) out[i] = 0.0f;
}

// ---------------------------------------------------------------------------
// Kernel 1: u = q @ k^T / 8  via V_WMMA_F32_16X16X4_F32, plus per-row max.
// Block = 256 threads = 8 waves; wave w computes 16x16 tile at
// (row0 = by*16, col0 = bx*128 + w*16). Grid = (M/128, N/16).
// ---------------------------------------------------------------------------
__global__ __launch_bounds__(256) void scores_kernel(const float* __restrict__ q,
                                                     const float* __restrict__ k,
                                                     float* __restrict__ attn,
                                                     unsigned int* __restrict__ rowmax_key) {
    const int lane = threadIdx.x & 31;
    const int wave = threadIdx.x >> 5;
    const int hi   = lane >> 4;     // half-wave select (K pairs {0,1} vs {2,3})
    const int l16  = lane & 15;
    const int row0 = blockIdx.y * 16;
    const int col0 = blockIdx.x * 128 + wave * 16;

    // A lane view: q[row0+l16][d0 + 2*hi + {0,1}]   (16x4 MxK layout)
    // B lane view: k[col0+l16][d0 + 2*hi + {0,1}]   (4x16 KxN layout, B[k][n]=K[n][k])
    const float* qrow = q + (size_t)(row0 + l16) * DIM + 2 * hi;
    const float* krow = k + (size_t)(col0 + l16) * DIM + 2 * hi;

    v8f acc = {0.f, 0.f, 0.f, 0.f, 0.f, 0.f, 0.f, 0.f};
#pragma unroll
    for (int d0 = 0; d0 < DIM; d0 += 4) {
        v2f a = *(const v2f*)(qrow + d0);
        v2f b = *(const v2f*)(krow + d0);
        acc = __builtin_amdgcn_wmma_f32_16x16x4_f32(false, a, false, b,
                                                    (short)0, acc, false, false);
    }
#pragma unroll
    for (int i = 0; i < 8; ++i) acc[i] *= INV_SCALE;

    // D layout: VGPR i -> row = row0 + i + 8*hi, col = col0 + l16
#pragma unroll
    for (int i = 0; i < 8; ++i) {
        int r = row0 + i + 8 * hi;
        attn[(size_t)r * M_K + col0 + l16] = acc[i];
    }
    // Row-max: reduce over the 16 lanes of each half (the N dimension).
#pragma unroll
    for (int i = 0; i < 8; ++i) {
        float m = acc[i];
        m = fmaxf(m, __shfl_xor(m, 1, 32));
        m = fmaxf(m, __shfl_xor(m, 2, 32));
        m = fmaxf(m, __shfl_xor(m, 4, 32));
        m = fmaxf(m, __shfl_xor(m, 8, 32));
        if (l16 == 0) {
            int r = row0 + i + 8 * hi;
            atomicMax(&rowmax_key[r], f2key(m));
        }
    }
}

// ---------------------------------------------------------------------------
// Kernel 2: e = exp(u - rowmax) in place; rowsum[row] += sum(e).
// Grid = (M/4096, N); block handles 1 row x 4096 cols with float4 I/O.
// ---------------------------------------------------------------------------
__global__ __launch_bounds__(256) void exp_kernel(float* __restrict__ attn,
                                                  const unsigned int* __restrict__ rowmax_key,
                                                  float* __restrict__ rowsum) {
    const int row = blockIdx.y;
    float* p = attn + (size_t)row * M_K + blockIdx.x * 4096;
    const float rmax = key2f(rowmax_key[row]);

    float partial = 0.0f;
    for (int i = threadIdx.x * 4; i < 4096; i += 256 * 4) {
        float4 t = *(float4*)(p + i);
        t.x = __expf(t.x - rmax);
        t.y = __expf(t.y - rmax);
        t.z = __expf(t.z - rmax);
        t.w = __expf(t.w - rmax);
        *(float4*)(p + i) = t;
        partial += t.x + t.y + t.z + t.w;
    }
    __shared__ float red[256];
    red[threadIdx.x] = partial;
    __syncthreads();
    for (int s = 128; s > 0; s >>= 1) {
        if ((int)threadIdx.x < s) red[threadIdx.x] += red[threadIdx.x + s];
        __syncthreads();
    }
    if (threadIdx.x == 0) atomicAdd(&rowsum[row], red[0]);
}

// ---------------------------------------------------------------------------
// Kernel 3: p = e / rowsum (written back) and output += p @ v via WMMA f32.
// Grid = (4, N/16). Block = 8 waves; wave w owns m-slice of 512 columns and
// all 4 d-tiles (D=64). Partials atomically accumulated into `out`.
// ---------------------------------------------------------------------------
__global__ __launch_bounds__(256) void norm_out_kernel(float* __restrict__ attn,
                                                       const float* __restrict__ v,
                                                       const float* __restrict__ rowsum,
                                                       float* __restrict__ out) {
    const int lane = threadIdx.x & 31;
    const int wave = threadIdx.x >> 5;
    const int hi   = lane >> 4;
    const int l16  = lane & 15;
    const int row0 = blockIdx.y * 16;
    const int m_begin = blockIdx.x * 4096 + wave * 512;

    const int arow = row0 + l16;                   // this lane's A-matrix row
    const float inv_sum = 1.0f / rowsum[arow];
    float* prow = attn + (size_t)arow * M_K;

    v8f acc0 = {0.f,0.f,0.f,0.f,0.f,0.f,0.f,0.f};
    v8f acc1 = acc0, acc2 = acc0, acc3 = acc0;

    for (int m0 = m_begin; m0 < m_begin + 512; m0 += 4) {
        // A: p[arow][m0 + 2*hi + {0,1}] — normalize and write p back
        // (each tile element touched exactly once across the wave).
        v2f a = *(v2f*)(prow + m0 + 2 * hi);
        a[0] *= inv_sum;
        a[1] *= inv_sum;
        *(v2f*)(prow + m0 + 2 * hi) = a;
        // B: b.x = v[m0+2*hi][d0+l16], b.y = v[m0+2*hi+1][d0+l16]
        const float* vb = v + (size_t)(m0 + 2 * hi) * DIM + l16;
        v2f b0 = { vb[0],  vb[DIM + 0]  };
        v2f b1 = { vb[16], vb[DIM + 16] };
        v2f b2 = { vb[32], vb[DIM + 32] };
        v2f b3 = { vb[48], vb[DIM + 48] };
        acc0 = __builtin_amdgcn_wmma_f32_16x16x4_f32(false, a, false, b0, (short)0, acc0, false, false);
        acc1 = __builtin_amdgcn_wmma_f32_16x16x4_f32(false, a, false, b1, (short)0, acc1, false, false);
        acc2 = __builtin_amdgcn_wmma_f32_16x16x4_f32(false, a, false, b2, (short)0, acc2, false, false);
        acc3 = __builtin_amdgcn_wmma_f32_16x16x4_f32(false, a, false, b3, (short)0, acc3, false, false);
    }
#pragma unroll
    for (int i = 0; i < 8; ++i) {
        int r = row0 + i + 8 * hi;
        atomicAdd(&out[(size_t)r * DIM +  0 + l16], acc0[i]);
        atomicAdd(&out[(size_t)r * DIM + 16 + l16], acc1[i]);
        atomicAdd(&out[(size_t)r * DIM + 32 + l16], acc2[i]);
        atomicAdd(&out[(size_t)r * DIM + 48 + l16], acc3[i]);
    }
}

// ---------------------------------------------------------------------------
// Kernel 4: exact top-160 per row, LDS-resident. The 64KB score row is staged
// into LDS (320KB/WGP on CDNA5) with GLOBAL_LOAD_ASYNC_TO_LDS_B128 (inline
// asm, ASYNCcnt-tracked); radix-select + selection then run entirely out of
// LDS. One 256-thread block per row; dynamic LDS sized at launch.
// ---------------------------------------------------------------------------
#define TOPK_SMEM_BYTES (M_K * sizeof(float)          /* rowbuf  */ \
                         + 256 * sizeof(unsigned int) /* hist    */ \
                         + KSEL * sizeof(int)         /* idxList */ \
                         + 4 * 64 * sizeof(float))    /* accum   */

__global__ __launch_bounds__(256) void topk_kernel(const float* __restrict__ attn,
                                                   const float* __restrict__ v,
                                                   float* __restrict__ sel) {
    extern __shared__ unsigned char smem_raw[];
    float*        rowbuf  = (float*)smem_raw;                  // [M_K]
    unsigned int* hist    = (unsigned int*)(rowbuf + M_K);     // [256]
    int*          idxList = (int*)(hist + 256);                // [KSEL]
    float*        accum   = (float*)(idxList + KSEL);          // [4][64]

    __shared__ unsigned int s_prefix, s_want, s_cnt, s_tie;

    const int row = blockIdx.x;
    const float* p = attn + (size_t)row * M_K;

    // ---- Stage the row into LDS ----
#if defined(__gfx1250__)
    {
        // Raw LDS byte offset of rowbuf (AS3 pointers are 32-bit LDS offsets).
        unsigned lds_base =
            (unsigned)(unsigned long long)(__attribute__((address_space(3))) char*)(void*)rowbuf;
        for (int j = threadIdx.x; j < M_K / 4; j += 256) {
            unsigned           ldst  = lds_base + (unsigned)j * 16u;
            unsigned long long gaddr = (unsigned long long)(p + 4 * j);
            // LDS byte address comes from the vdst VGPR; GV mode: 64-bit vaddr, no saddr.
            asm volatile("global_load_async_to_lds_b128 %0, %1, off"
                         :
                         : "v"(ldst), "v"(gaddr)
                         : "memory");
        }
        asm volatile("s_wait_asynccnt 0x0" ::: "memory");
    }
#else
    for (int j = threadIdx.x; j < M_K / 4; j += 256)
        ((float4*)rowbuf)[j] = ((const float4*)p)[j];
#endif
    if (threadIdx.x == 0) { s_prefix = 0u; s_want = KSEL; }
    __syncthreads();

    // ---- 4-pass (8 bits each) radix select for the exact KSEL-th largest key.
    for (int pass = 0; pass < 4; ++pass) {
        const int shift = 24 - pass * 8;
        hist[threadIdx.x] = 0u;
        __syncthreads();
        const unsigned int prefix = s_prefix;
        const unsigned int mask = (pass == 0) ? 0u : (0xFFFFFFFFu << (shift + 8));
        for (int i = threadIdx.x; i < M_K; i += 256) {
            unsigned int key = f2key(rowbuf[i]);
            if ((key & mask) == prefix)
                atomicAdd(&hist[(key >> shift) & 0xFFu], 1u);
        }
        __syncthreads();
        if (threadIdx.x == 0) {
            unsigned int want = s_want, cum = 0;
            int b = 255;
            for (; b > 0; --b) {
                unsigned int c = hist[b];
                if (cum + c >= want) break;
                cum += c;
            }
            s_prefix = prefix | ((unsigned int)b << shift);
            s_want = want - cum;   // rank within the selected bin
        }
        __syncthreads();
    }

    const unsigned int kth = s_prefix;          // exact 160th-largest key
    const unsigned int ties_allowed = s_want;   // # of ==kth entries to keep
    if (threadIdx.x == 0) { s_cnt = 0u; s_tie = 0u; }
    __syncthreads();

    for (int i = threadIdx.x; i < M_K; i += 256) {
        unsigned int key = f2key(rowbuf[i]);
        if (key > kth) {
            idxList[atomicAdd(&s_cnt, 1u)] = i;
        } else if (key == kth) {
            unsigned int t = atomicAdd(&s_tie, 1u);
            if (t < ties_allowed) idxList[atomicAdd(&s_cnt, 1u)] = i;
        }
    }
    __syncthreads();

    // ---- Mean of the 160 selected v rows: 4 groups x 64 dims.
    const int g = threadIdx.x >> 6;
    const int d = threadIdx.x & 63;
    float sum = 0.0f;
    for (int j = g; j < KSEL; j += 4)
        sum += v[(size_t)idxList[j] * DIM + d];
    accum[g * 64 + d] = sum;
    __syncthreads();
    if (threadIdx.x < 64) {
        float s = accum[0 * 64 + threadIdx.x] + accum[1 * 64 + threadIdx.x] +
                  accum[2 * 64 + threadIdx.x] + accum[3 * 64 + threadIdx.x];
        sel[(size_t)row * DIM + threadIdx.x] = s * (1.0f / (float)KSEL);
    }
}

// ---------------------------------------------------------------------------
extern "C" void kernel_launch(void* const* d_in, const int* in_sizes, int n_in,
                              void* d_out, int out_size, void* d_ws, size_t ws_size,
                              hipStream_t stream) {
    (void)in_sizes; (void)n_in; (void)out_size; (void)ws_size;
    const float* q = (const float*)d_in[0];
    const float* k = (const float*)d_in[1];
    const float* v = (const float*)d_in[2];
    // d_in[3] is topk==10; KSEL = 10*16 baked in.

    float* attn = (float*)d_out;                         // [N, M]
    float* out  = attn + (size_t)N_Q * M_K;              // [N, D]
    float* sel  = out  + (size_t)N_Q * DIM;              // [N, D]

    unsigned int* rowmax_key = (unsigned int*)d_ws;      // [N]
    float*        rowsum     = (float*)(rowmax_key + N_Q); // [N]

    // Allow >64KB dynamic LDS (CDNA5 WGP has 320KB); idempotent, capture-safe.
    (void)hipFuncSetAttribute((const void*)topk_kernel,
                              hipFuncAttributeMaxDynamicSharedMemorySize,
                              (int)TOPK_SMEM_BYTES);

    init_kernel<<<(N_Q * DIM + 255) / 256, 256, 0, stream>>>(rowmax_key, rowsum, out);

    dim3 g1(M_K / 128, N_Q / 16);    // (128, 512)
    scores_kernel<<<g1, 256, 0, stream>>>(q, k, attn, rowmax_key);

    dim3 g2(M_K / 4096, N_Q);        // (4, 8192)
    exp_kernel<<<g2, 256, 0, stream>>>(attn, rowmax_key, rowsum);

    dim3 g3(4, N_Q / 16);            // (4, 512)
    norm_out_kernel<<<g3, 256, 0, stream>>>(attn, v, rowsum, out);

    topk_kernel<<<N_Q, 256, TOPK_SMEM_BYTES, stream>>>(attn, v, sel);
}